// ImplicitSNN_61125974557352
// MI455X (gfx1250) — compile-verified
//
#include <hip/hip_runtime.h>
#include <math.h>

// ---------------- problem constants ----------------
#define SEQ     8192
#define DM      1024      // d_model
#define DS      16        // d_state
#define DL      32        // d_latent
#define ITERS   20
#define TOLSQ   1e-10f    // (1e-5)^2

#define NCHUNK  16        // SEQ / 512
#define CHUNK   512
#define NSEG    32        // segments per chunk (16 elems each)

typedef float v2f __attribute__((ext_vector_type(2)));
typedef float v8f __attribute__((ext_vector_type(8)));

__device__ __forceinline__ v8f wmma_f32x4(v2f a, v2f b, v8f c) {
    // D = A(16x4 f32) * B(4x16 f32) + C(16x16 f32)
    return __builtin_amdgcn_wmma_f32_16x16x4_f32(
        /*neg_a=*/false, a, /*neg_b=*/false, b,
        /*c_mod=*/(short)0, c, /*reuse_a=*/false, /*reuse_b=*/false);
}

__device__ __forceinline__ float sigmoidf_(float x) {
    return 1.0f / (1.0f + __expf(-x));
}

// ---------------- init: z=0, partial norms, done flag ----------------
__global__ void ImplicitSNN_init_kernel(float* __restrict__ z,
                                        float* __restrict__ partial,
                                        int* __restrict__ done) {
    int idx = blockIdx.x * 256 + threadIdx.x;
    if (idx < SEQ * DL) z[idx] = 0.0f;
    if (idx < 512) partial[idx] = (idx == 0) ? 3.0e38f : 0.0f; // z_prev=inf -> first norm huge
    if (idx == 0) *done = 0;
}

// ---------------- fused embedding-gather + 4-way projection GEMM (WMMA) ----
// P[row][0:16)=x@Wlam_x^T+blam  [16:32)=Wu  [32:48)=Wf  [48:64)=Wf2
// block: 128 threads (4 waves), 16 rows per block; each wave one 16-col group.
__global__ __launch_bounds__(128)
void ImplicitSNN_proj_kernel(const int* __restrict__ tokens,
                             const float* __restrict__ emb,
                             const float* __restrict__ Wlam, const float* __restrict__ blam,
                             const float* __restrict__ Wu,   const float* __restrict__ bu,
                             const float* __restrict__ Wf,   const float* __restrict__ bf,
                             const float* __restrict__ Wf2,  const float* __restrict__ bf2,
                             float* __restrict__ P) {
    __shared__ float xt[16 * DM];  // transposed + xor-swizzled x tile: x[r][k] at xt[k*16 + ((k&15)^r)]
    const int tid = threadIdx.x;
    const int rowbase = blockIdx.x * 16;

    for (int idx = tid; idx < 16 * DM; idx += 128) {
        int r = idx >> 10, k = idx & (DM - 1);
        long ta = (long)tokens[rowbase + r] * DM;
        xt[k * 16 + ((k & 15) ^ r)] = emb[ta + k];
    }
    __syncthreads();

    const int wave = tid >> 5, lane = tid & 31;
    const int half = lane >> 4, mn = lane & 15;

    const float* W; const float* bias; int off, ld;
    if (wave == 0)      { W = Wlam; bias = blam; off = DL; ld = DL + DM; }
    else if (wave == 1) { W = Wu;   bias = bu;   off = DL; ld = DL + DM; }
    else if (wave == 2) { W = Wf;   bias = bf;   off = DS; ld = DS + DM; }
    else                { W = Wf2;  bias = bf2;  off = DS; ld = DS + DM; }
    const float* wrow = W + (long)mn * ld + off;   // B[k][n=mn] = W[mn][off+k]

    v8f c = {0.f,0.f,0.f,0.f,0.f,0.f,0.f,0.f};
#pragma unroll 4
    for (int k = 0; k < DM; k += 4) {
        int ka = k + half * 2;
        v2f a, b;
        a.x = xt[ka * 16       + ((ka & 15) ^ mn)];
        a.y = xt[(ka + 1) * 16 + (((ka + 1) & 15) ^ mn)];
        b.x = wrow[ka];
        b.y = wrow[ka + 1];
        c = wmma_f32x4(a, b, c);
    }
    float bv = bias[mn];
#pragma unroll
    for (int i = 0; i < 8; ++i) {
        int row = rowbase + i + 8 * half;
        P[(long)row * 64 + wave * 16 + mn] = c[i] + bv;
    }
}

// ---------------- sticky convergence check (deterministic) ----------------
__global__ void ImplicitSNN_check_kernel(const float* __restrict__ partial,
                                         int* __restrict__ done) {
    if (threadIdx.x == 0 && blockIdx.x == 0) {
        float s = 0.0f;
        for (int i = 0; i < 512; ++i) s += partial[i];
        if (s < TOLSQ) *done = 1;
    }
}

// ---------------- lam/u from z (K=32, tiny -> VALU) ----------------
__global__ __launch_bounds__(256)
void ImplicitSNN_lamu_kernel(const float* __restrict__ z,
                             const float* __restrict__ Wlam,
                             const float* __restrict__ Wu,
                             const float* __restrict__ P,
                             float* __restrict__ lam, float* __restrict__ u) {
    int idx = blockIdx.x * 256 + threadIdx.x;    // SEQ*16
    int row = idx >> 4, j = idx & 15;
    const float* zr = z + (long)row * DL;
    const float* wl = Wlam + (long)j * (DL + DM);
    const float* wu = Wu   + (long)j * (DL + DM);
    float aL = P[(long)row * 64 + j];
    float aU = P[(long)row * 64 + 16 + j];
#pragma unroll
    for (int k = 0; k < DL; ++k) {
        float zv = zr[k];
        aL += zv * wl[k];
        aU += zv * wu[k];
    }
    lam[idx] = sigmoidf_(aL);
    u[idx]   = aU;
}

// ---------------- scan phase A: segment compose + per-chunk segment scan ---
// binop(left,right) = (a1*a2, a2*b1 + b2), left = earlier in sequence.
__global__ __launch_bounds__(512)
void ImplicitSNN_scanA_kernel(const float* __restrict__ lam, const float* __restrict__ u,
                              float* __restrict__ segA, float* __restrict__ segB,
                              float* __restrict__ chA,  float* __restrict__ chB) {
    __shared__ float sA[512], sB[512];
    int tid = threadIdx.x;
    int c = tid & 15, s = tid >> 4;      // channel, segment
    int g = blockIdx.x;                  // chunk
    int r0 = g * CHUNK + s * 16;
    float A = 1.0f, B = 0.0f;
#pragma unroll
    for (int e = 0; e < 16; ++e) {
        float l  = lam[(long)(r0 + e) * 16 + c];
        float uu = u  [(long)(r0 + e) * 16 + c];
        A = A * l;
        B = l * B + uu;
    }
    sA[tid] = A; sB[tid] = B;
    __syncthreads();
    if (s == 0) {   // threads 0..15: serial exclusive scan over 32 segments
        float pA = 1.0f, pB = 0.0f;
        for (int ss = 0; ss < NSEG; ++ss) {
            segA[(long)(g * NSEG + ss) * 16 + c] = pA;
            segB[(long)(g * NSEG + ss) * 16 + c] = pB;
            float aA = sA[ss * 16 + c], aB = sB[ss * 16 + c];
            float nB = aA * pB + aB;
            pA = pA * aA;
            pB = nB;
        }
        chA[g * 16 + c] = pA;
        chB[g * 16 + c] = pB;
    }
}

// ---------------- scan phase B: exclusive scan over chunk aggregates ------
__global__ void ImplicitSNN_scanB_kernel(const float* __restrict__ chA, const float* __restrict__ chB,
                                         float* __restrict__ cpA, float* __restrict__ cpB) {
    int c = threadIdx.x;
    if (c >= 16) return;
    float pA = 1.0f, pB = 0.0f;
    for (int g = 0; g < NCHUNK; ++g) {
        cpA[g * 16 + c] = pA;
        cpB[g * 16 + c] = pB;
        float aA = chA[g * 16 + c], aB = chB[g * 16 + c];
        float nB = aA * pB + aB;
        pA = pA * aA;
        pB = nB;
    }
}

// ---------------- scan phase C: materialize h (exclusive->h_prev or inclusive)
__global__ __launch_bounds__(512)
void ImplicitSNN_scanC_kernel(const float* __restrict__ lam, const float* __restrict__ u,
                              const float* __restrict__ segA, const float* __restrict__ segB,
                              const float* __restrict__ cpA,  const float* __restrict__ cpB,
                              float* __restrict__ hout, int inclusive) {
    int tid = threadIdx.x;
    int c = tid & 15, s = tid >> 4;
    int g = blockIdx.x;
    float cAv = cpA[g * 16 + c], cBv = cpB[g * 16 + c];
    float sAp = segA[(long)(g * NSEG + s) * 16 + c];
    float sBp = segB[(long)(g * NSEG + s) * 16 + c];
    // compose(chunk-prefix, seg-prefix): h before this segment (h_{-1}=0 => h = B)
    float hrun = sAp * cBv + sBp;
    int r0 = g * CHUNK + s * 16;
#pragma unroll
    for (int e = 0; e < 16; ++e) {
        int row = r0 + e;
        if (!inclusive) hout[(long)row * 16 + c] = hrun;   // h_prev[row] = h_{row-1}
        float l  = lam[(long)row * 16 + c];
        float uu = u  [(long)row * 16 + c];
        hrun = l * hrun + uu;
        if (inclusive) hout[(long)row * 16 + c] = hrun;    // h[row]
    }
}

// ---------------- f_theta + deterministic ||dz||^2 partials ---------------
__global__ __launch_bounds__(256)
void ImplicitSNN_ftheta_kernel(float* __restrict__ z,
                               const float* __restrict__ h,
                               const float* __restrict__ Wf,
                               const float* __restrict__ Wf2,
                               const float* __restrict__ P,
                               float* __restrict__ partial,
                               const int* __restrict__ done) {
    __shared__ float red[256];
    int tid = threadIdx.x;
    int idx = blockIdx.x * 256 + tid;    // SEQ*16
    int row = idx >> 4, j = idx & 15;
    int frozen = *done;
    float dsq = 0.0f;
    if (!frozen) {
        const float* hp  = h   + (long)row * DS;
        const float* wf  = Wf  + (long)j * (DS + DM);
        const float* wf2 = Wf2 + (long)j * (DS + DM);
        float aA = P[(long)row * 64 + 32 + j];
        float aS = P[(long)row * 64 + 48 + j];
#pragma unroll
        for (int k = 0; k < DS; ++k) {
            float hv = hp[k];
            aA += hv * wf[k];
            aS += hv * wf2[k];
        }
        float alpha = sigmoidf_(aA);
        float sigma = sigmoidf_(aS);
        float uu = z[(long)row * DL + j];
        float vv = z[(long)row * DL + 16 + j];
        float th = tanhf(30.0f * (vv - uu));
        float du =  1.0f - alpha * __expf(15.6f * vv) * (1.0f - 0.26f * (0.3f - uu)) + sigma * th;
        float dv = -1.0f + alpha * __expf(15.6f * uu) * (1.0f + 0.26f * (0.3f - vv)) + sigma * th;
        float nu = uu + 0.001f * du;
        float nv = vv + 0.001f * dv;
        z[(long)row * DL + j]      = nu;
        z[(long)row * DL + 16 + j] = nv;
        float d1 = nu - uu, d2 = nv - vv;
        dsq = d1 * d1 + d2 * d2;
    }
    red[tid] = dsq;
    __syncthreads();
    for (int st = 128; st > 0; st >>= 1) {
        if (tid < st) red[tid] += red[tid + st];
        __syncthreads();
    }
    if (tid == 0) partial[blockIdx.x] = red[0];
}

// ---------------- output GEMM: out = [z|h] @ Wout^T + bout (WMMA, K=48) ---
__global__ __launch_bounds__(128)
void ImplicitSNN_outgemm_kernel(const float* __restrict__ z,
                                const float* __restrict__ h,
                                const float* __restrict__ Wout,
                                const float* __restrict__ bout,
                                float* __restrict__ out) {
    const int tid = threadIdx.x;
    const int wave = tid >> 5, lane = tid & 31;
    const int half = lane >> 4, mn = lane & 15;
    const int rowTile = blockIdx.x >> 4;                       // 512 row tiles
    const int colTile = ((blockIdx.x & 15) << 2) + wave;       // 64 col tiles
    const int rowbase = rowTile * 16;
    const int n = colTile * 16 + mn;

    const float* wr = Wout + (long)n * (DL + DS);
    const float* zr = z + (long)(rowbase + mn) * DL;
    const float* hr = h + (long)(rowbase + mn) * DS;

    v8f c = {0.f,0.f,0.f,0.f,0.f,0.f,0.f,0.f};
#pragma unroll
    for (int k = 0; k < DL + DS; k += 4) {
        int ka = k + half * 2;
        v2f a, b;
        a.x = (ka     < DL) ? zr[ka]     : hr[ka - DL];
        a.y = (ka + 1 < DL) ? zr[ka + 1] : hr[ka + 1 - DL];
        b.x = wr[ka];
        b.y = wr[ka + 1];
        c = wmma_f32x4(a, b, c);
    }
    float bv = bout[n];
#pragma unroll
    for (int i = 0; i < 8; ++i)
        out[(long)(rowbase + i + 8 * half) * DM + n] = c[i] + bv;
}

// ---------------- host launch ----------------
extern "C" void kernel_launch(void* const* d_in, const int* in_sizes, int n_in,
                              void* d_out, int out_size, void* d_ws, size_t ws_size,
                              hipStream_t stream) {
    const int*   tokens = (const int*)  d_in[0];
    const float* emb    = (const float*)d_in[1];
    const float* Wf     = (const float*)d_in[2];
    const float* bf     = (const float*)d_in[3];
    const float* Wf2    = (const float*)d_in[4];
    const float* bf2    = (const float*)d_in[5];
    const float* Wlam   = (const float*)d_in[6];
    const float* blam   = (const float*)d_in[7];
    const float* Wu     = (const float*)d_in[8];
    const float* bu     = (const float*)d_in[9];
    const float* Wout   = (const float*)d_in[10];
    const float* bout   = (const float*)d_in[11];
    float* out = (float*)d_out;

    float* w = (float*)d_ws;
    float* P    = w;                 // SEQ*64
    float* z    = P    + SEQ * 64;   // SEQ*32
    float* lam  = z    + SEQ * DL;   // SEQ*16
    float* u    = lam  + SEQ * DS;   // SEQ*16
    float* h    = u    + SEQ * DS;   // SEQ*16
    float* segA = h    + SEQ * DS;   // NCHUNK*NSEG*16
    float* segB = segA + NCHUNK * NSEG * 16;
    float* chA  = segB + NCHUNK * NSEG * 16;  // NCHUNK*16
    float* chB  = chA  + NCHUNK * 16;
    float* cpA  = chB  + NCHUNK * 16;
    float* cpB  = cpA  + NCHUNK * 16;
    float* partial = cpB + NCHUNK * 16;       // 512
    int*   done = (int*)(partial + 512);

    ImplicitSNN_init_kernel<<<1024, 256, 0, stream>>>(z, partial, done);
    ImplicitSNN_proj_kernel<<<SEQ / 16, 128, 0, stream>>>(
        tokens, emb, Wlam, blam, Wu, bu, Wf, bf, Wf2, bf2, P);

    for (int it = 0; it < ITERS; ++it) {
        ImplicitSNN_check_kernel<<<1, 32, 0, stream>>>(partial, done);
        ImplicitSNN_lamu_kernel<<<SEQ * 16 / 256, 256, 0, stream>>>(z, Wlam, Wu, P, lam, u);
        ImplicitSNN_scanA_kernel<<<NCHUNK, 512, 0, stream>>>(lam, u, segA, segB, chA, chB);
        ImplicitSNN_scanB_kernel<<<1, 32, 0, stream>>>(chA, chB, cpA, cpB);
        ImplicitSNN_scanC_kernel<<<NCHUNK, 512, 0, stream>>>(lam, u, segA, segB, cpA, cpB, h, 0);
        ImplicitSNN_ftheta_kernel<<<SEQ * 16 / 256, 256, 0, stream>>>(z, h, Wf, Wf2, P, partial, done);
    }

    // final lam/u + inclusive scan for h_star, then output GEMM
    ImplicitSNN_lamu_kernel<<<SEQ * 16 / 256, 256, 0, stream>>>(z, Wlam, Wu, P, lam, u);
    ImplicitSNN_scanA_kernel<<<NCHUNK, 512, 0, stream>>>(lam, u, segA, segB, chA, chB);
    ImplicitSNN_scanB_kernel<<<1, 32, 0, stream>>>(chA, chB, cpA, cpB);
    ImplicitSNN_scanC_kernel<<<NCHUNK, 512, 0, stream>>>(lam, u, segA, segB, cpA, cpB, h, 1);
    ImplicitSNN_outgemm_kernel<<<(SEQ / 16) * 16, 128, 0, stream>>>(z, h, Wout, bout, out);
}